// MixHop_82231443849284
// MI455X (gfx1250) — compile-verified
//
#include <hip/hip_runtime.h>
#include <hip/hip_bf16.h>

typedef __attribute__((ext_vector_type(2))) float v2f;
typedef __attribute__((ext_vector_type(8))) float v8f;

#define DIN 128
#define HCAT 384
#define DOUT 64

// ---------------- degree / normalization ----------------

__global__ void deg_init_kernel(float* __restrict__ deg, int N) {
    int i = blockIdx.x * blockDim.x + threadIdx.x;
    if (i < N) deg[i] = 1.0f;   // self loop contributes 1
}

__global__ void deg_edges_kernel(const long long* __restrict__ ei, float* __restrict__ deg, int E) {
    int e = blockIdx.x * blockDim.x + threadIdx.x;
    if (e < E) {
        long long c = ei[(size_t)E + e];   // col = destination
        atomicAdd(&deg[c], 1.0f);
    }
}

__global__ void make_dis_kernel(float* __restrict__ deg, int N) {
    int i = blockIdx.x * blockDim.x + threadIdx.x;
    if (i < N) deg[i] = rsqrtf(deg[i]);    // deg >= 1 always (self loops)
}

// ---------------- SpMM: dst = Ahat * src ----------------
// self-loop term also initializes dst:  dst[n] = src[n] * dis[n]^2
__global__ void prop_self_kernel(const float* __restrict__ src, const float* __restrict__ dis,
                                 float* __restrict__ dst, int N) {
    int idx = blockIdx.x * blockDim.x + threadIdx.x;   // N*32 threads
    int n  = idx >> 5;
    int f4 = (idx & 31) * 4;
    if (n < N) {
        float s = dis[n] * dis[n];
        const float4 v = *(const float4*)(src + (size_t)n * DIN + f4);
        float4 o; o.x = v.x * s; o.y = v.y * s; o.z = v.z * s; o.w = v.w * s;
        *(float4*)(dst + (size_t)n * DIN + f4) = o;
    }
}

// edge term: dst[col] += src[row] * dis[row]*dis[col]
__global__ void prop_edges_kernel(const long long* __restrict__ ei, const float* __restrict__ src,
                                  const float* __restrict__ dis, float* __restrict__ dst, int E) {
    int idx = blockIdx.x * blockDim.x + threadIdx.x;   // E*32 threads
    int e  = idx >> 5;
    if (e >= E) return;
    int f4 = (idx & 31) * 4;
    long long r = ei[e];
    long long c = ei[(size_t)E + e];
    float nrm = dis[r] * dis[c];
    const float4 v = *(const float4*)(src + (size_t)r * DIN + f4);
    float* p = dst + (size_t)c * DIN + f4;
    atomicAdd(p + 0, v.x * nrm);
    atomicAdd(p + 1, v.y * nrm);
    atomicAdd(p + 2, v.z * nrm);
    atomicAdd(p + 3, v.w * nrm);
}

// ---------------- hidden GEMMs via WMMA f32 16x16x4 ----------------
// One wave computes a 16x16 tile of h_part = A_part @ W_part^T + b_part, ReLU, stored
// into h[N,384] at column offset part*128.
__global__ void __launch_bounds__(256) gemm_hidden_kernel(
    const float* __restrict__ x,  const float* __restrict__ ax, const float* __restrict__ a2x,
    const float* __restrict__ W0, const float* __restrict__ W1, const float* __restrict__ W2,
    const float* __restrict__ b0, const float* __restrict__ b1, const float* __restrict__ b2,
    float* __restrict__ h, int N, int Mtiles)
{
    int wave = blockIdx.x * 8 + (threadIdx.x >> 5);
    int lane = threadIdx.x & 31;
    int mt   = wave / 24;           // waves in a block share the M tile (A reuse in L2/L0)
    if (mt >= Mtiles) return;       // wave-uniform: EXEC stays all-1s for WMMA
    int rem  = wave % 24;
    int part = rem >> 3;            // 0,1,2 -> x/ax/a2x with W0/W1/W2
    int nt   = rem & 7;             // 8 column tiles of 16 (HOP=128)

    const float* A    = (part == 0) ? x  : (part == 1) ? ax : a2x;
    const float* W    = (part == 0) ? W0 : (part == 1) ? W1 : W2;
    const float* bias = (part == 0) ? b0 : (part == 1) ? b1 : b2;

    int row0  = mt * 16;
    int sub   = lane & 15;          // M index for A, N index for B/D
    int khalf = lane >> 4;          // 0: K=0,1  1: K=2,3 (per ISA A/B f32 layout)

    int arow = row0 + sub;  if (arow >= N) arow = N - 1;          // clamp (N%16==0 normally)
    const float* aptr = A + (size_t)arow * DIN + khalf * 2;
    const float* wptr = W + (size_t)(nt * 16 + sub) * DIN + khalf * 2;  // B(k,n) = W[n,k]

    v8f c = {0.f, 0.f, 0.f, 0.f, 0.f, 0.f, 0.f, 0.f};
#pragma unroll
    for (int k = 0; k < DIN; k += 4) {
        v2f a = *(const v2f*)(aptr + k);
        v2f b = *(const v2f*)(wptr + k);
        c = __builtin_amdgcn_wmma_f32_16x16x4_f32(false, a, false, b, (short)0, c, false, false);
    }

    // D layout: VGPR r -> M = r + 8*khalf, N = sub
    float bv   = bias[nt * 16 + sub];
    int   mb   = row0 + khalf * 8;
    int   coff = part * 128 + nt * 16 + sub;
#pragma unroll
    for (int r = 0; r < 8; ++r) {
        int m = mb + r;
        if (m < N) {
            float v = c[r] + bv;
            h[(size_t)m * HCAT + coff] = v > 0.f ? v : 0.f;
        }
    }
}

// ---------------- output GEMM: out = relu_h @ Wout^T + bout ----------------
__global__ void __launch_bounds__(256) gemm_out_kernel(
    const float* __restrict__ h, const float* __restrict__ Wout, const float* __restrict__ bout,
    float* __restrict__ out, int N, int Mtiles)
{
    int wave = blockIdx.x * 8 + (threadIdx.x >> 5);
    int lane = threadIdx.x & 31;
    int mt   = wave >> 2;
    if (mt >= Mtiles) return;
    int nt   = wave & 3;            // 4 column tiles of 16 (DOUT=64)

    int row0  = mt * 16;
    int sub   = lane & 15;
    int khalf = lane >> 4;

    int arow = row0 + sub;  if (arow >= N) arow = N - 1;
    const float* aptr = h    + (size_t)arow * HCAT + khalf * 2;
    const float* wptr = Wout + (size_t)(nt * 16 + sub) * HCAT + khalf * 2;

    v8f c = {0.f, 0.f, 0.f, 0.f, 0.f, 0.f, 0.f, 0.f};
#pragma unroll 8
    for (int k = 0; k < HCAT; k += 4) {
        v2f a = *(const v2f*)(aptr + k);
        v2f b = *(const v2f*)(wptr + k);
        c = __builtin_amdgcn_wmma_f32_16x16x4_f32(false, a, false, b, (short)0, c, false, false);
    }

    float bv = bout[nt * 16 + sub];
    int   mb = row0 + khalf * 8;
#pragma unroll
    for (int r = 0; r < 8; ++r) {
        int m = mb + r;
        if (m < N)
            out[(size_t)m * DOUT + nt * 16 + sub] = c[r] + bv;
    }
}

// ---------------- launcher ----------------

extern "C" void kernel_launch(void* const* d_in, const int* in_sizes, int n_in,
                              void* d_out, int out_size, void* d_ws, size_t ws_size,
                              hipStream_t stream) {
    const float*     x    = (const float*)d_in[0];
    const long long* ei   = (const long long*)d_in[1];   // int64 [2,E]
    const float*     W0   = (const float*)d_in[2];
    const float*     b0   = (const float*)d_in[3];
    const float*     W1   = (const float*)d_in[4];
    const float*     b1   = (const float*)d_in[5];
    const float*     W2   = (const float*)d_in[6];
    const float*     b2   = (const float*)d_in[7];
    const float*     Wout = (const float*)d_in[8];
    const float*     bout = (const float*)d_in[9];
    float*           out  = (float*)d_out;

    const int N = in_sizes[0] / DIN;
    const int E = in_sizes[1] / 2;

    // workspace layout
    float* deg  = (float*)d_ws;                    // N floats; becomes dis in place
    float* ax   = deg + N;                         // N*128
    float* a2x  = ax  + (size_t)N * DIN;           // N*128
    float* hbuf = a2x + (size_t)N * DIN;           // N*384

    const int T = 256;

    // degree + D^{-1/2}
    deg_init_kernel<<<(N + T - 1) / T, T, 0, stream>>>(deg, N);
    deg_edges_kernel<<<(E + T - 1) / T, T, 0, stream>>>(ei, deg, E);
    make_dis_kernel<<<(N + T - 1) / T, T, 0, stream>>>(deg, N);

    // ax = Ahat x ; a2x = Ahat ax
    long long nthN = (long long)N * 32;
    long long nthE = (long long)E * 32;
    prop_self_kernel <<<(int)((nthN + T - 1) / T), T, 0, stream>>>(x, deg, ax, N);
    prop_edges_kernel<<<(int)((nthE + T - 1) / T), T, 0, stream>>>(ei, x, deg, ax, E);
    prop_self_kernel <<<(int)((nthN + T - 1) / T), T, 0, stream>>>(ax, deg, a2x, N);
    prop_edges_kernel<<<(int)((nthE + T - 1) / T), T, 0, stream>>>(ei, ax, deg, a2x, E);

    // hidden GEMMs + bias + ReLU -> hbuf[N,384]
    const int Mtiles = (N + 15) / 16;
    {
        long long waves = (long long)Mtiles * 24;         // 3 parts * 8 col tiles
        int blocks = (int)((waves + 7) / 8);
        gemm_hidden_kernel<<<blocks, 256, 0, stream>>>(x, ax, a2x, W0, W1, W2,
                                                       b0, b1, b2, hbuf, N, Mtiles);
    }
    // output GEMM -> out[N,64]
    {
        long long waves = (long long)Mtiles * 4;          // 4 col tiles of 16
        int blocks = (int)((waves + 7) / 8);
        gemm_out_kernel<<<blocks, 256, 0, stream>>>(hbuf, Wout, bout, out, N, Mtiles);
    }
}